// NeuraLogicLayer_86818468922051
// MI455X (gfx1250) — compile-verified
//
#include <hip/hip_runtime.h>
#include <hip/hip_bf16.h>
#include <stdint.h>

#define TPB 256

// -------------------- init: x = e0, out = 0 (vectorized) --------------------
__global__ void nl_init(float4* __restrict__ x4, float4* __restrict__ out4,
                        float* __restrict__ x, float* __restrict__ out, int N) {
    const int N4 = N >> 2;
    int i = blockIdx.x * blockDim.x + threadIdx.x;
    if (i < N4) {
        float4 z; z.x = 0.f; z.y = 0.f; z.z = 0.f; z.w = 0.f;
        out4[i] = z;
        if (i == 0) z.x = 1.0f;     // x[0] = 1
        x4[i] = z;
    }
    // scalar tail (N % 4; N=2M has none, stay generic)
    int t = (N4 << 2) + i;
    if (i < (N - (N4 << 2))) { x[t] = 0.f; out[t] = 0.f; }
}

// -------------------- edge scatter kernel --------------------
// Streams src[] through LDS via double-buffered async global->LDS copies
// (16B per lane, contiguous 512B per wave per issue, NT temporal hint).
// Gathers x[src]; only when nonzero (x is extremely sparse: <= ~8^layer
// nonzeros out of 2M) touches dst/widx and issues a HW float atomic.
__global__ __launch_bounds__(TPB) void nl_edges(
    const int*   __restrict__ src,
    const int*   __restrict__ dst,
    const int*   __restrict__ widx,
    const float* __restrict__ weights,
    const float* __restrict__ x,
    float*       __restrict__ out,
    int E) {
    __shared__ int4 stage[2][TPB];

    const int tid    = threadIdx.x;
    const int gtid   = blockIdx.x * TPB + tid;
    const int stride = gridDim.x * TPB;
    const int E4     = E >> 2;   // groups of 4 edges

    // Low 32 bits of a generic shared-memory pointer == LDS byte address.
    const uint32_t lds0 = (uint32_t)(uintptr_t)(&stage[0][tid]);
    const uint32_t lds1 = (uint32_t)(uintptr_t)(&stage[1][tid]);

    int i   = gtid;
    int buf = 0;

    if (i < E4) {
        const int* gp = src + (size_t)4 * i;
        asm volatile("global_load_async_to_lds_b128 %0, %1, off th:TH_LOAD_NT"
                     :: "v"(lds0), "v"(gp) : "memory");
    }

    while (i < E4) {
        const int  inext = i + stride;
        const bool more  = inext < E4;
        if (more) {
            const int*     gp = src + (size_t)4 * inext;
            const uint32_t ln = buf ? lds0 : lds1;   // fill the *other* buffer
            asm volatile("global_load_async_to_lds_b128 %0, %1, off th:TH_LOAD_NT"
                         :: "v"(ln), "v"(gp) : "memory");
            asm volatile("s_wait_asynccnt 1" ::: "memory");  // current tile done
        } else {
            asm volatile("s_wait_asynccnt 0" ::: "memory");
        }

        const int4 sv   = stage[buf][tid];
        const int  base = 4 * i;
        const int  ss[4] = {sv.x, sv.y, sv.z, sv.w};
#pragma unroll
        for (int k = 0; k < 4; ++k) {
            const float v = x[ss[k]];
            if (v != 0.0f) {                       // wave-coherent skip
                const int d  = __builtin_nontemporal_load(dst + base + k);
                const int wi = __builtin_nontemporal_load(widx + base + k);
                unsafeAtomicAdd(out + d, v * weights[wi]);  // global_atomic_add_f32
            }
        }
        buf ^= 1;
        i = inext;
    }

    // Scalar tail for E not divisible by 4 (E=16M is, but stay correct).
    for (int e = (E4 << 2) + gtid; e < E; e += stride) {
        const float v = x[src[e]];
        if (v != 0.0f) {
            unsafeAtomicAdd(out + dst[e], v * weights[widx[e]]);
        }
    }
}

// -------------- residual ReLU + accumulator reset (vectorized) --------------
__global__ void nl_relu_res(float4* __restrict__ x4, float4* __restrict__ out4,
                            float* __restrict__ x, float* __restrict__ out, int N) {
    const int N4 = N >> 2;
    int i = blockIdx.x * blockDim.x + threadIdx.x;
    if (i < N4) {
        float4 o = out4[i];
        float4 v = x4[i];
        v.x += (o.x > 0.f) ? o.x : 0.f;
        v.y += (o.y > 0.f) ? o.y : 0.f;
        v.z += (o.z > 0.f) ? o.z : 0.f;
        v.w += (o.w > 0.f) ? o.w : 0.f;
        x4[i] = v;
        float4 z; z.x = 0.f; z.y = 0.f; z.z = 0.f; z.w = 0.f;
        out4[i] = z;                 // ready for next layer
    }
    int t = (N4 << 2) + i;
    if (i < (N - (N4 << 2))) {
        const float o = out[t];
        x[t] += (o > 0.f) ? o : 0.f;
        out[t] = 0.f;
    }
}

// ---- final: fuse last layer's residual ReLU (only element 0 is needed) ----
__global__ void nl_write(const float* __restrict__ x, const float* __restrict__ out,
                         float* __restrict__ r) {
    const float o = out[0];
    r[0] = x[0] + ((o > 0.f) ? o : 0.f);
}

// -------------------- host launch --------------------
extern "C" void kernel_launch(void* const* d_in, const int* in_sizes, int n_in,
                              void* d_out, int out_size, void* d_ws, size_t ws_size,
                              hipStream_t stream) {
    const float* weights = (const float*)d_in[0];
    const int*   src     = (const int*)d_in[1];
    const int*   dst     = (const int*)d_in[2];
    const int*   widx    = (const int*)d_in[3];

    const int  L = 4;
    const int  N = 2000000;                 // fixed by the reference
    const long long Etot = in_sizes[1];
    const int  E = (int)(Etot / L);

    float* x   = (float*)d_ws;              // N floats
    float* out = x + N;                     // N floats  (16 MB total in d_ws)
    float* res = (float*)d_out;

    const int N4       = N >> 2;
    const int nBlocks4 = (N4 + TPB - 1) / TPB;

    nl_init<<<nBlocks4, TPB, 0, stream>>>((float4*)x, (float4*)out, x, out, N);

    int edgeBlocks = ((E >> 2) + TPB - 1) / TPB;
    if (edgeBlocks > 2048) edgeBlocks = 2048;   // grid-stride, ~8 tiles/thread
    if (edgeBlocks < 1) edgeBlocks = 1;

    for (int l = 0; l < L; ++l) {
        const size_t off = (size_t)l * (size_t)E;
        nl_edges<<<edgeBlocks, TPB, 0, stream>>>(src + off, dst + off, widx + off,
                                                 weights, x, out, E);
        if (l < L - 1) {    // last layer's residual is fused into nl_write
            nl_relu_res<<<nBlocks4, TPB, 0, stream>>>((float4*)x, (float4*)out,
                                                      x, out, N);
        }
    }

    nl_write<<<1, 1, 0, stream>>>(x, out, res);
}